// TopKMoE_46737834115362
// MI455X (gfx1250) — compile-verified
//
#include <hip/hip_runtime.h>
#include <math.h>

#define T_TOK   8192
#define D_MODEL 1024
#define F_FFN   4096
#define N_EXP   8
#define CAP     2048
#define N_ASG   (T_TOK * 2)
#define OUT_ELEMS (T_TOK * D_MODEL)

typedef float v2f __attribute__((ext_vector_type(2)));
typedef float v8f __attribute__((ext_vector_type(8)));

// ---------------------------------------------------------------- init
__global__ void moe_init_kernel(float* importance, int* load_cnt) {
  int i = threadIdx.x;
  if (i < N_EXP) { importance[i] = 0.0f; load_cnt[i] = 0; }
}

// ---------------------------------------------------------------- router
// one wave (32 lanes) per token: 1024-dot with 8 router columns,
// butterfly reduce, softmax, top-2 (stable tie-break), gates, atomics.
__global__ __launch_bounds__(256) void moe_router_kernel(
    const float* __restrict__ xf, const float* __restrict__ wr,
    float* importance, int* load_cnt, int* e_arr, float* gate_arr) {
  const int lane = threadIdx.x & 31;
  const int wave = threadIdx.x >> 5;
  const int t = blockIdx.x * 8 + wave;
  const float* xr = xf + (size_t)t * D_MODEL;

  float acc[N_EXP];
#pragma unroll
  for (int e = 0; e < N_EXP; ++e) acc[e] = 0.0f;
  for (int j = 0; j < D_MODEL / 32; ++j) {
    int i = j * 32 + lane;
    float xv = xr[i];
    const float* w = wr + i * N_EXP;
#pragma unroll
    for (int e = 0; e < N_EXP; ++e) acc[e] += xv * w[e];
  }
#pragma unroll
  for (int off = 16; off > 0; off >>= 1) {
#pragma unroll
    for (int e = 0; e < N_EXP; ++e)
      acc[e] += __shfl_xor(acc[e], off, 32);
  }
  if (lane == 0) {
    float m = acc[0];
#pragma unroll
    for (int e = 1; e < N_EXP; ++e) m = fmaxf(m, acc[e]);
    float p[N_EXP], s = 0.0f;
#pragma unroll
    for (int e = 0; e < N_EXP; ++e) { p[e] = expf(acc[e] - m); s += p[e]; }
    float inv = 1.0f / s;
#pragma unroll
    for (int e = 0; e < N_EXP; ++e) p[e] *= inv;
    int t1 = 0;
#pragma unroll
    for (int e = 1; e < N_EXP; ++e) if (p[e] > p[t1]) t1 = e;   // lowest index wins ties
    int t2 = (t1 == 0) ? 1 : 0;
#pragma unroll
    for (int e = 0; e < N_EXP; ++e)
      if (e != t1 && p[e] > p[t2]) t2 = e;
    float denom = fmaxf(p[t1] + p[t2], 1e-9f);
    e_arr[2 * t]     = t1;
    e_arr[2 * t + 1] = t2;
    gate_arr[2 * t]     = p[t1] / denom;
    gate_arr[2 * t + 1] = p[t2] / denom;
#pragma unroll
    for (int e = 0; e < N_EXP; ++e) atomicAdd(&importance[e], p[e]);
    atomicAdd(&load_cnt[t1], 1);
  }
}

// ---------------------------------------------------------------- stable rank
__global__ void moe_hist_kernel(const int* __restrict__ e_arr, int* blockhist) {
  __shared__ int h[N_EXP];
  int tid = threadIdx.x;
  if (tid < N_EXP) h[tid] = 0;
  __syncthreads();
  int n = blockIdx.x * 256 + tid;
  atomicAdd(&h[e_arr[n]], 1);
  __syncthreads();
  if (tid < N_EXP) blockhist[blockIdx.x * N_EXP + tid] = h[tid];
}

__global__ void moe_scan_kernel(const int* __restrict__ blockhist, int* prefix) {
  int e = threadIdx.x;
  if (e < N_EXP) {
    int run = 0;
    for (int b = 0; b < N_ASG / 256; ++b) {
      prefix[b * N_EXP + e] = run;
      run += blockhist[b * N_EXP + e];
    }
  }
}

__global__ void moe_place_kernel(const int* __restrict__ e_arr,
                                 const int* __restrict__ prefix, int* a2s) {
  __shared__ int se[256];
  int tid = threadIdx.x;
  int n = blockIdx.x * 256 + tid;
  int e = e_arr[n];
  se[tid] = e;
  __syncthreads();
  int r = 0;
  for (int m = 0; m < tid; ++m) r += (se[m] == e) ? 1 : 0;  // stable in-block rank
  int pos = prefix[blockIdx.x * N_EXP + e] + r;
  a2s[n] = (pos < CAP) ? (e * CAP + pos) : -1;
}

// ---------------------------------------------------------------- scatter
__global__ __launch_bounds__(256) void moe_scatter_kernel(
    const float* __restrict__ xf, const int* __restrict__ a2s, float* xbuf) {
  int n = blockIdx.x;
  int slot = a2s[n];
  if (slot < 0) return;
  int tok = n >> 1;
  const float4* src = (const float4*)(xf + (size_t)tok * D_MODEL);
  float4* dst = (float4*)(xbuf + (size_t)slot * D_MODEL);
  dst[threadIdx.x] = src[threadIdx.x];
}

// ---------------------------------------------------------------- WMMA GEMM
// C[M,N] = A[M,K] * B[K,N], all row-major.  MODE 0: C = acc.
// MODE 1: C = silu(C) * acc  (fused SwiGLU epilogue; C holds h, acc is u).
//
// Double-buffered LDS staged with GLOBAL_LOAD_ASYNC_TO_LDS_B128 (ASYNCcnt).
// A and B tiles are stored with 16B-chunk XOR swizizzles (conflict-free
// fragment reads) since async copies land contiguous 16B chunks.
#define BM 128
#define BN 128
#define BK 32

template <int MODE>
__global__ __launch_bounds__(256) void moe_gemm_kernel(
    const float* __restrict__ A, const float* __restrict__ B,
    float* __restrict__ C, int M, int N, int K) {
  __shared__ float As[2][BM * BK];   // 16 KB per stage, chunk-swizzled
  __shared__ float Bs[2][BK * BN];   // 16 KB per stage, chunk-swizzled

  const int tid  = threadIdx.x;
  const int lane = tid & 31;
  const int wave = tid >> 5;
  const int wm = wave >> 2;       // 0..1 : 64-row band
  const int wn = wave & 3;        // 0..3 : 32-col band
  const int lhalf = lane >> 4;    // 0/1 -> K pair select (A/B frag layout)
  const int l16 = lane & 15;
  const int m0 = blockIdx.y * BM;
  const int n0 = blockIdx.x * BN;

  v8f acc[4][2];
#pragma unroll
  for (int s = 0; s < 4; ++s)
#pragma unroll
    for (int t = 0; t < 2; ++t)
      acc[s][t] = (v8f){0.f, 0.f, 0.f, 0.f, 0.f, 0.f, 0.f, 0.f};

  // issue 8 async B128 copies (4 for A tile, 4 for B tile) into stage buf
  auto issue_stage = [&](int k0, int buf) {
#pragma unroll
    for (int i = 0; i < 4; ++i) {           // A tile: 128 x 32
      int q = i * 256 + tid;
      int r = q >> 3, c = (q & 7) * 4;
      int cc = (c >> 2) ^ ((r >> 1) & 7);   // chunk XOR swizzle
      const float* g = A + (size_t)(m0 + r) * K + k0 + c;
      unsigned l = (unsigned)(uintptr_t)&As[buf][r * BK + (cc << 2)];
      asm volatile("global_load_async_to_lds_b128 %0, %1, off"
                   :: "v"(l), "v"(g) : "memory");
    }
#pragma unroll
    for (int i = 0; i < 4; ++i) {           // B tile: 32 x 128
      int q = i * 256 + tid;
      int r = q >> 5, c = (q & 31) * 4;
      int cb = (c >> 2) ^ (((r >> 1) & 1) << 3);  // chunk XOR swizzle
      const float* g = B + (size_t)(k0 + r) * N + n0 + c;
      unsigned l = (unsigned)(uintptr_t)&Bs[buf][r * BN + (cb << 2)];
      asm volatile("global_load_async_to_lds_b128 %0, %1, off"
                   :: "v"(l), "v"(g) : "memory");
    }
  };

  const int nstage = K / BK;
  issue_stage(0, 0);

  for (int st = 0; st < nstage; ++st) {
    const int buf = st & 1;
    if (st + 1 < nstage) {
      issue_stage((st + 1) * BK, buf ^ 1);
      asm volatile("s_wait_asynccnt 0x8" ::: "memory");   // stage st landed
    } else {
      asm volatile("s_wait_asynccnt 0x0" ::: "memory");
    }
    __syncthreads();

#pragma unroll
    for (int kk = 0; kk < BK; kk += 4) {
      v2f a[4], b[2];
#pragma unroll
      for (int s = 0; s < 4; ++s) {
        int row = wm * 64 + s * 16 + l16;
        int cc = (kk >> 2) ^ ((row >> 1) & 7);
        a[s] = *(const v2f*)(&As[buf][row * BK + (cc << 2) + lhalf * 2]);
      }
#pragma unroll
      for (int t = 0; t < 2; ++t) {
        int col = wn * 32 + t * 16 + l16;
        int krow = kk + lhalf * 2;
        int swz0 = ((col >> 2) ^ ((((krow)     >> 1) & 1) << 3)) * 4 + (col & 3);
        int swz1 = ((col >> 2) ^ ((((krow + 1) >> 1) & 1) << 3)) * 4 + (col & 3);
        b[t][0] = Bs[buf][krow * BN + swz0];
        b[t][1] = Bs[buf][(krow + 1) * BN + swz1];
      }
#pragma unroll
      for (int s = 0; s < 4; ++s)
#pragma unroll
        for (int t = 0; t < 2; ++t)
          acc[s][t] = __builtin_amdgcn_wmma_f32_16x16x4_f32(
              false, a[s], false, b[t], (short)0, acc[s][t], false, false);
    }
    __syncthreads();
  }

  // epilogue: C/D frag layout -> VGPR v, lanes<16: M=v, lanes>=16: M=v+8
#pragma unroll
  for (int s = 0; s < 4; ++s) {
#pragma unroll
    for (int t = 0; t < 2; ++t) {
      int gc  = n0 + wn * 32 + t * 16 + l16;
      int gr0 = m0 + wm * 64 + s * 16 + lhalf * 8;
#pragma unroll
      for (int v = 0; v < 8; ++v) {
        size_t idx = (size_t)(gr0 + v) * N + gc;
        float val = acc[s][t][v];
        if (MODE == 0) {
          C[idx] = val;
        } else {
          float h = C[idx];
          C[idx] = (h / (1.0f + expf(-h))) * val;   // silu(h) * u
        }
      }
    }
  }
}

// ---------------------------------------------------------------- gather
__global__ __launch_bounds__(256) void moe_gather_kernel(
    const float* __restrict__ ybuf, const int* __restrict__ a2s,
    const float* __restrict__ gate_arr, float* __restrict__ out) {
  int t = blockIdx.x;
  int s0 = a2s[2 * t], s1 = a2s[2 * t + 1];
  float g0 = gate_arr[2 * t], g1 = gate_arr[2 * t + 1];
  int i = threadIdx.x;
  float4 r = {0.f, 0.f, 0.f, 0.f};
  if (s0 >= 0) {
    float4 y = ((const float4*)(ybuf + (size_t)s0 * D_MODEL))[i];
    r.x += g0 * y.x; r.y += g0 * y.y; r.z += g0 * y.z; r.w += g0 * y.w;
  }
  if (s1 >= 0) {
    float4 y = ((const float4*)(ybuf + (size_t)s1 * D_MODEL))[i];
    r.x += g1 * y.x; r.y += g1 * y.y; r.z += g1 * y.z; r.w += g1 * y.w;
  }
  ((float4*)(out + (size_t)t * D_MODEL))[i] = r;
}

// ---------------------------------------------------------------- aux loss
__global__ void moe_aux_kernel(const float* importance, const int* load_cnt,
                               float* out_aux) {
  if (threadIdx.x == 0) {
    float lb = 0.0f;
    for (int e = 0; e < N_EXP; ++e)
      lb += (importance[e] / (float)T_TOK) * ((float)load_cnt[e] / (float)T_TOK);
    out_aux[0] = 0.01f * lb * (float)N_EXP;
  }
}

// ---------------------------------------------------------------- launch
extern "C" void kernel_launch(void* const* d_in, const int* in_sizes, int n_in,
                              void* d_out, int out_size, void* d_ws, size_t ws_size,
                              hipStream_t stream) {
  const float* x  = (const float*)d_in[0];
  const float* wr = (const float*)d_in[1];
  const float* w1 = (const float*)d_in[2];
  const float* w3 = (const float*)d_in[3];
  const float* w2 = (const float*)d_in[4];
  float* out = (float*)d_out;

  char* p = (char*)d_ws;
  float* importance = (float*)p;                       // 8 f32
  int*   load_cnt   = (int*)(p + 32);                  // 8 i32
  int*   blockhist  = (int*)(p + 64);                  // 64*8 i32
  int*   prefix     = (int*)(p + 64 + 2048);           // 64*8 i32
  int*   e_arr      = (int*)(p + 64 + 4096);           // 16384 i32
  float* gate_arr   = (float*)(p + 64 + 4096 + 65536); // 16384 f32
  int*   a2s        = (int*)(p + 64 + 4096 + 131072);  // 16384 i32
  float* xbuf   = (float*)(p + 262144);                           // 8*2048*1024 f32
  float* actbuf = xbuf + (size_t)N_EXP * CAP * D_MODEL;           // 2048*4096 f32
  float* ybuf   = actbuf + (size_t)CAP * F_FFN;                   // 8*2048*1024 f32

  moe_init_kernel<<<1, 32, 0, stream>>>(importance, load_cnt);
  moe_router_kernel<<<T_TOK / 8, 256, 0, stream>>>(x, wr, importance, load_cnt,
                                                   e_arr, gate_arr);
  moe_hist_kernel<<<N_ASG / 256, 256, 0, stream>>>(e_arr, blockhist);
  moe_scan_kernel<<<1, 32, 0, stream>>>(blockhist, prefix);
  moe_place_kernel<<<N_ASG / 256, 256, 0, stream>>>(e_arr, prefix, a2s);
  moe_scatter_kernel<<<N_ASG, 256, 0, stream>>>(x, a2s, xbuf);

  dim3 g1(F_FFN / BN, CAP / BM);    // 32 x 16
  dim3 g2(D_MODEL / BN, CAP / BM);  // 8 x 16
  for (int e = 0; e < N_EXP; ++e) {
    const float* xe  = xbuf + (size_t)e * CAP * D_MODEL;
    const float* w1e = w1 + (size_t)e * D_MODEL * F_FFN;
    const float* w3e = w3 + (size_t)e * D_MODEL * F_FFN;
    const float* w2e = w2 + (size_t)e * F_FFN * D_MODEL;
    float* ye = ybuf + (size_t)e * CAP * D_MODEL;
    moe_gemm_kernel<0><<<g1, 256, 0, stream>>>(xe, w1e, actbuf, CAP, F_FFN, D_MODEL);
    moe_gemm_kernel<1><<<g1, 256, 0, stream>>>(xe, w3e, actbuf, CAP, F_FFN, D_MODEL);
    moe_gemm_kernel<0><<<g2, 256, 0, stream>>>(actbuf, w2e, ye, CAP, D_MODEL, F_FFN);
  }

  moe_gather_kernel<<<T_TOK, 256, 0, stream>>>(ybuf, a2s, gate_arr, out);
  moe_aux_kernel<<<1, 32, 0, stream>>>(importance, load_cnt, out + OUT_ELEMS);
}